// DrugTargetGCN_38594576122355
// MI455X (gfx1250) — compile-verified
//
#include <hip/hip_runtime.h>
#include <hip/hip_bf16.h>
#include <math.h>

typedef __attribute__((ext_vector_type(2))) float v2f;
typedef __attribute__((ext_vector_type(8))) float v8f;

// ---------------- Kernel 1: init degree with self-loop weight (1.0) ----------------
__global__ void k_init_deg(float* __restrict__ deg, int n) {
    int i = blockIdx.x * blockDim.x + threadIdx.x;
    if (i < n) deg[i] = 1.0f;
}

// ---------------- Kernel 2: degree accumulation at dst ----------------
__global__ void k_deg(const long long* __restrict__ dst, const float* __restrict__ w,
                      float* __restrict__ deg, int E) {
    int e = blockIdx.x * blockDim.x + threadIdx.x;
    if (e < E) {
        int d = (int)dst[e];
        atomicAdd(&deg[d], w[e]);
    }
}

// ---------------- Kernel 3: dis = rsqrt(deg) in place; seed agg with self-loop ----
__global__ void k_dis_selfagg(float* __restrict__ deg_dis, const float* __restrict__ x,
                              float* __restrict__ agg, int n) {
    int i = blockIdx.x * blockDim.x + threadIdx.x;
    if (i < n) {
        float dg = deg_dis[i];
        float r  = dg > 0.0f ? rsqrtf(dg) : 0.0f;
        deg_dis[i] = r;                  // dis overwrites deg
        float s = r * r;                 // self-loop norm = dis[i]*1*dis[i]
        agg[2 * i + 0] = s * x[2 * i + 0];
        agg[2 * i + 1] = s * x[2 * i + 1];
    }
}

// ---------------- Kernel 4: edge aggregation of 2-dim raw features -----------------
__global__ void k_agg_edges(const long long* __restrict__ src, const long long* __restrict__ dst,
                            const float* __restrict__ w, const float* __restrict__ dis,
                            const float* __restrict__ x, float* __restrict__ agg, int E) {
    int e = blockIdx.x * blockDim.x + threadIdx.x;
    if (e < E) {
        int s = (int)src[e];
        int d = (int)dst[e];
        float nm = dis[s] * w[e] * dis[d];
        atomicAdd(&agg[2 * d + 0], nm * x[2 * s + 0]);
        atomicAdd(&agg[2 * d + 1], nm * x[2 * s + 1]);
    }
}

// ---------------- Kernel 5: fused node MLP via V_WMMA_F32_16X16X4_F32 --------------
// Per wave: one 16-node tile. A = agg tile (16x4, K=2 padded), B = W1 col-tile (4x16).
// 8 WMMAs cover 128 features; then z+b1 -> ELU -> dot W2 -> h2; out = b2 + dis^2*h2.
__global__ __launch_bounds__(256)
void k_node_mlp(const float* __restrict__ agg, const float* __restrict__ dis,
                const float* __restrict__ W1, const float* __restrict__ b1,
                const float* __restrict__ W2, const float* __restrict__ b2,
                float* __restrict__ h2, float* __restrict__ out, int n_tiles, int N) {
    int wave = threadIdx.x >> 5;
    int lane = threadIdx.x & 31;
    int tile = blockIdx.x * 8 + wave;
    if (tile >= n_tiles) return;           // wave-uniform exit; active waves keep EXEC=all-1s

    int node_base = tile * 16;
    int m  = lane & 15;
    bool hi = lane >= 16;

    // A matrix (16x4 f32): lanes 0-15 hold M=lane, VGPR0=K0, VGPR1=K1;
    // lanes 16-31 hold K2,K3 which are the zero-padding of K=2.
    int node = node_base + m;
    int nclamp = node < N ? node : (N - 1);
    float a0 = agg[2 * nclamp + 0];
    float a1 = agg[2 * nclamp + 1];
    v2f a;
    a.x = hi ? 0.0f : a0;
    a.y = hi ? 0.0f : a1;

    float acc[8];
#pragma unroll
    for (int v = 0; v < 8; ++v) acc[v] = 0.0f;

#pragma unroll
    for (int t = 0; t < 8; ++t) {
        int j  = t * 16 + m;               // output feature column for this lane
        // B matrix (4x16 f32): lanes 0-15 hold N=lane, VGPR0=K0 row of W1, VGPR1=K1 row;
        // lanes 16-31 (K2,K3) are zero padding.
        float w1a = W1[j];                  // W1[0, j]
        float w1b = W1[128 + j];            // W1[1, j]
        v2f b;
        b.x = hi ? 0.0f : w1a;
        b.y = hi ? 0.0f : w1b;

        v8f c = {};
        c = __builtin_amdgcn_wmma_f32_16x16x4_f32(
                /*neg_a=*/false, a, /*neg_b=*/false, b,
                /*c_mod=*/(short)0, c, /*reuse_a=*/false, /*reuse_b=*/false);

        float b1v = b1[j];
        float w2v = W2[j];
        // C/D layout: lane holds rows M = v + (hi?8:0), column N = m, for v = 0..7
#pragma unroll
        for (int v = 0; v < 8; ++v) {
            float z = c[v] + b1v;
            float e = z > 0.0f ? z : expm1f(z);   // ELU(alpha=1)
            acc[v] += e * w2v;
        }
    }

    // Reduce over the 16 columns held by the 16 lanes of each half-wave.
#pragma unroll
    for (int off = 1; off < 16; off <<= 1) {
#pragma unroll
        for (int v = 0; v < 8; ++v) acc[v] += __shfl_xor(acc[v], off, 32);
    }

    if (m == 0) {
        float b2v = b2[0];
        int rbase = node_base + (hi ? 8 : 0);
#pragma unroll
        for (int v = 0; v < 8; ++v) {
            int r = rbase + v;
            if (r < N) {
                float hv = acc[v];
                float dv = dis[r];
                h2[r]  = hv;
                out[r] = b2v + dv * dv * hv;   // bias + layer-2 self-loop term
            }
        }
    }
}

// ---------------- Kernel 6: layer-2 scalar edge scatter ----------------------------
__global__ void k_out_edges(const long long* __restrict__ src, const long long* __restrict__ dst,
                            const float* __restrict__ w, const float* __restrict__ dis,
                            const float* __restrict__ h2, float* __restrict__ out, int E) {
    int e = blockIdx.x * blockDim.x + threadIdx.x;
    if (e < E) {
        int s = (int)src[e];
        int d = (int)dst[e];
        float nm = dis[s] * w[e] * dis[d];
        atomicAdd(&out[d], nm * h2[s]);
    }
}

extern "C" void kernel_launch(void* const* d_in, const int* in_sizes, int n_in,
                              void* d_out, int out_size, void* d_ws, size_t ws_size,
                              hipStream_t stream) {
    const float*     x   = (const float*)d_in[0];       // [N,2]
    const long long* ei  = (const long long*)d_in[1];   // [2,E] int64
    const float*     ew  = (const float*)d_in[2];       // [E]
    const float*     W1  = (const float*)d_in[3];       // [2,128]
    const float*     b1  = (const float*)d_in[4];       // [128]
    const float*     W2  = (const float*)d_in[5];       // [128,1]
    const float*     b2  = (const float*)d_in[6];       // [1]

    const int N = in_sizes[0] / 2;
    const int E = in_sizes[2];
    const long long* srcI = ei;
    const long long* dstI = ei + E;

    // Workspace layout (floats): [0,N) deg->dis, [N,3N) agg, [3N,4N) h2
    float* ws  = (float*)d_ws;
    float* dis = ws;            // deg, then dis in-place
    float* agg = ws + N;
    float* h2  = ws + 3 * (size_t)N;
    float* out = (float*)d_out;

    const int TB = 256;
    const int gN = (N + TB - 1) / TB;
    const int gE = (E + TB - 1) / TB;
    const int n_tiles = (N + 15) / 16;
    const int gT = (n_tiles + 7) / 8;     // 8 waves (tiles) per 256-thread block

    k_init_deg  <<<gN, TB, 0, stream>>>(dis, N);
    k_deg       <<<gE, TB, 0, stream>>>(dstI, ew, dis, E);
    k_dis_selfagg<<<gN, TB, 0, stream>>>(dis, x, agg, N);
    k_agg_edges <<<gE, TB, 0, stream>>>(srcI, dstI, ew, dis, x, agg, E);
    k_node_mlp  <<<gT, TB, 0, stream>>>(agg, dis, W1, b1, W2, b2, h2, out, n_tiles, N);
    k_out_edges <<<gE, TB, 0, stream>>>(srcI, dstI, ew, dis, h2, out, E);
}